// AggregateStgcn_70557722739211
// MI455X (gfx1250) — compile-verified
//
#include <hip/hip_runtime.h>
#include <hip/hip_bf16.h>

typedef __attribute__((ext_vector_type(2))) float v2f;
typedef __attribute__((ext_vector_type(8))) float v8f;

#define V_NODES 8192
#define C_CH 4
#define K_KS 4
#define F_FIFO 16
#define NSPLIT 8
#define VSPLIT (V_NODES / NSPLIT)   // 1024 rows per v-split
#define COLS_PER_BLOCK 64           // 4 waves x 16 columns
#define THREADS_K2 128

// ---------------------------------------------------------------------------
// Kernel 1: fold x (1, C*K, 1, V) over the C axis -> s[k][v], k=0..3
// s[k][v] = sum_c x[(c*K + k)*V + v]
// ---------------------------------------------------------------------------
__global__ void stgcn_prep(const float* __restrict__ x, float* __restrict__ s) {
    int v = blockIdx.x * blockDim.x + threadIdx.x;
    if (v < V_NODES) {
#pragma unroll
        for (int k = 0; k < K_KS; ++k) {
            float acc = 0.f;
#pragma unroll
            for (int c = 0; c < C_CH; ++c)
                acc += x[(c * K_KS + k) * V_NODES + v];
            s[k * V_NODES + v] = acc;
        }
    }
}

// ---------------------------------------------------------------------------
// Kernel 2: split-K  (4x8192) @ (8192x8192) via V_WMMA_F32_16X16X4_F32.
// Block: 128 threads = 4 waves; wave handles a 16-column tile of the output.
// Grid: (V/64 column blocks, NSPLIT v-splits).
// A tile (16x4 f32): rows 0..3 = s[k][v..v+3], rows 4..15 replicate mod 4
//   (their D rows are never stored, so replication avoids divergence and
//   keeps EXEC all-ones as WMMA requires).
// B tile (4x16 f32): graph[v+kb + {0,1}][w0 + n].
// ---------------------------------------------------------------------------
__global__ void __launch_bounds__(THREADS_K2)
stgcn_wmma(const float* __restrict__ graph, const float* __restrict__ s,
           float* __restrict__ partial) {
    __shared__ float s_lds[K_KS * VSPLIT];   // 16 KB: s slice for this v-split

    const int tid   = threadIdx.x;
    const int lane  = tid & 31;
    const int wave  = tid >> 5;
    const int vbase = blockIdx.y * VSPLIT;

    // Cooperative, coalesced load of the s slice into LDS.
    for (int i = tid; i < K_KS * VSPLIT; i += THREADS_K2) {
        int k  = i >> 10;            // i / VSPLIT
        int vo = i & (VSPLIT - 1);
        s_lds[i] = s[k * V_NODES + vbase + vo];
    }
    __syncthreads();

    const int w0   = blockIdx.x * COLS_PER_BLOCK + wave * 16;
    const int n    = lane & 15;          // B column within the tile
    const int kb   = (lane >> 4) << 1;   // K pair base: 0 (lanes 0-15), 2 (16-31)
    const int mLds = lane & 3;           // A row, replicated mod 4

    const float* gp     = graph + (size_t)(vbase + kb) * V_NODES + (w0 + n);
    const float* a_base = &s_lds[mLds * VSPLIT + kb];

    v8f acc = {};
#pragma unroll 4
    for (int vo = 0; vo < VSPLIT; vo += 4) {
        v2f a = *(const v2f*)(a_base + vo);              // ds_load_b64
        v2f b;
        b.x = gp[0];                                     // graph[v+kb  ][w0+n]
        b.y = gp[V_NODES];                               // graph[v+kb+1][w0+n]
        __builtin_prefetch(gp + (size_t)32 * V_NODES, 0, 0);
        acc = __builtin_amdgcn_wmma_f32_16x16x4_f32(
            false, a, false, b, (short)0, acc, false, false);
        gp += (size_t)4 * V_NODES;
    }

    // D rows 0..3 live in acc[0..3], lanes 0..15 (N = lane).
    if (lane < 16) {
        float* p = partial + (size_t)blockIdx.y * (K_KS * V_NODES) + w0 + lane;
#pragma unroll
        for (int r = 0; r < 4; ++r)
            p[r * V_NODES] = acc[r];
    }
}

// ---------------------------------------------------------------------------
// Kernel 3: out[k][w] = sum_splits partial[sp][k][w]
//                     + sum_{j in 1,3,...,13} fifo[j][k][w]
// (new FIFO slot 0 = x_sum; even slots 2..14 of the pushed FIFO are old
//  frames 1,3,...,13)
// ---------------------------------------------------------------------------
__global__ void stgcn_finish(const float* __restrict__ fifo,
                             const float* __restrict__ partial,
                             float* __restrict__ out) {
    int i = blockIdx.x * blockDim.x + threadIdx.x;   // over C*V
    if (i >= C_CH * V_NODES) return;
    int k = i >> 13;            // i / V
    int w = i & (V_NODES - 1);

    float acc = 0.f;
#pragma unroll
    for (int j = 1; j <= 13; j += 2)
        acc += fifo[(j * C_CH + k) * V_NODES + w];
#pragma unroll
    for (int sp = 0; sp < NSPLIT; ++sp)
        acc += partial[(size_t)sp * (K_KS * V_NODES) + k * V_NODES + w];
    out[i] = acc;
}

// ---------------------------------------------------------------------------
extern "C" void kernel_launch(void* const* d_in, const int* in_sizes, int n_in,
                              void* d_out, int out_size, void* d_ws, size_t ws_size,
                              hipStream_t stream) {
    const float* x     = (const float*)d_in[0];   // (1, C*K, 1, V)
    const float* graph = (const float*)d_in[1];   // (V, V)
    const float* fifo  = (const float*)d_in[2];   // (1, F, C, V)
    // d_in[3] = stride (==2, baked into stgcn_finish)

    float* out     = (float*)d_out;               // (1, C, V, 1) -> [k*V + w]
    float* s_ws    = (float*)d_ws;                        // 4*V floats
    float* partial = s_ws + (size_t)K_KS * V_NODES;       // NSPLIT*4*V floats

    stgcn_prep<<<V_NODES / 256, 256, 0, stream>>>(x, s_ws);

    dim3 grid2(V_NODES / COLS_PER_BLOCK, NSPLIT);
    stgcn_wmma<<<grid2, THREADS_K2, 0, stream>>>(graph, s_ws, partial);

    stgcn_finish<<<(C_CH * V_NODES + 255) / 256, 256, 0, stream>>>(fifo, partial, out);
}